// SparseLinear_30915174597238
// MI455X (gfx1250) — compile-verified
//
#include <hip/hip_runtime.h>

typedef __attribute__((ext_vector_type(2))) float v2f;
typedef __attribute__((ext_vector_type(8))) float v8f;

namespace {
constexpr int kD     = 128;  // embedding dim
constexpr int kS     = 200;  // bag size
constexpr int kWaves = 8;    // waves per block (wave32)
}

// out[b,d] = sum_s w[b,s] * table[idx[b,s], d]
// One wave per batch row. Weighted sum done on the matrix pipe:
//   A (16x4 f32) = chunk weights broadcast down M
//   B (4x16 f32) = 4 gathered table rows x 16-wide d-tile
//   C (16x16 f32) accumulates across 50 s-chunks; every M row identical.
__global__ __launch_bounds__(256)
void sparse_embed_bag_wmma(const int* __restrict__ idxp,
                           const float* __restrict__ wp,
                           const float* __restrict__ tbl,
                           float* __restrict__ out,
                           int nBatch) {
  const int lane = threadIdx.x & 31;
  const int wave = threadIdx.x >> 5;
  int b = blockIdx.x * kWaves + wave;
  b = __builtin_amdgcn_readfirstlane(b);   // scalarize index/weight streams
  if (b >= nBatch) return;                 // wave-uniform exit: EXEC stays all-1s

  const int half = lane >> 4;              // 0: lanes 0-15, 1: lanes 16-31
  const int col  = lane & 15;              // N within a 16-wide d-tile
  const long long sbase = (long long)b * kS;

  v8f acc[8] = {};                         // 8 d-tiles x (16x16 f32 C layout)

  for (int s0 = 0; s0 < kS; s0 += 4) {
    // wave-uniform scalar fetches (s_load path)
    const int   i0 = idxp[sbase + s0 + 0];
    const int   i1 = idxp[sbase + s0 + 1];
    const int   i2 = idxp[sbase + s0 + 2];
    const int   i3 = idxp[sbase + s0 + 3];
    const float w0 = wp[sbase + s0 + 0];
    const float w1 = wp[sbase + s0 + 1];
    const float w2 = wp[sbase + s0 + 2];
    const float w3 = wp[sbase + s0 + 3];

    // Prefetch next chunk's 4 rows (4 x 512B): 32 lanes x 64B cachelines.
    if (s0 + 4 < kS) {
      const int pi = idxp[sbase + s0 + 4 + (lane >> 3)];
      const char* pp = (const char*)(tbl + (long long)pi * kD) + (lane & 7) * 64;
      __builtin_prefetch(pp, 0, 3);        // -> global_prefetch_b8
    }

    // A 16x4 f32 layout: VGPR0 = K0 | K2, VGPR1 = K1 | K3 ; A[m,k] = w[k]
    v2f A;
    A.x = half ? w2 : w0;
    A.y = half ? w3 : w1;

    // B 4x16 f32 layout: VGPR0 = row K0 | row K2, VGPR1 = row K1 | row K3.
    // The per-lane b32 gather *is* the B-operand build (64B coalesced runs).
    const float* r_even = tbl + (long long)(half ? i2 : i0) * kD + col;
    const float* r_odd  = tbl + (long long)(half ? i3 : i1) * kD + col;

#pragma unroll
    for (int t = 0; t < 8; ++t) {
      v2f Bm;
      Bm.x = r_even[t * 16];
      Bm.y = r_odd[t * 16];
      acc[t] = __builtin_amdgcn_wmma_f32_16x16x4_f32(
          /*neg_a=*/false, A, /*neg_b=*/false, Bm,
          /*c_mod=*/(short)0, acc[t], /*reuse_a=*/false, /*reuse_b=*/false);
    }
  }

  // All 16 M rows of each C tile are identical; emit row M=0 (VGPR0, lanes 0-15).
  if (lane < 16) {
#pragma unroll
    for (int t = 0; t < 8; ++t)
      out[(long long)b * kD + t * 16 + lane] = acc[t][0];
  }
}

extern "C" void kernel_launch(void* const* d_in, const int* in_sizes, int n_in,
                              void* d_out, int out_size, void* d_ws, size_t ws_size,
                              hipStream_t stream) {
  const int*   index  = (const int*)d_in[0];   // [B,S]
  const float* weight = (const float*)d_in[1]; // [B,S]
  const float* table  = (const float*)d_in[2]; // [V,D]
  float*       out    = (float*)d_out;         // [B,D]

  const int nBatch = out_size / kD;            // 4096
  dim3 block(32 * kWaves);                     // 256 threads = 8 waves
  dim3 grid((nBatch + kWaves - 1) / kWaves);   // 512 blocks
  sparse_embed_bag_wmma<<<grid, block, 0, stream>>>(index, weight, table, out, nBatch);
}